// PointDownsample_6176162972235
// MI455X (gfx1250) — compile-verified
//
#include <hip/hip_runtime.h>

typedef __attribute__((ext_vector_type(2))) float v2f;
typedef __attribute__((ext_vector_type(8))) float v8f;

#define N_PTS   8192
#define BSZ     4
#define M_CHILD 4096
#define KNN     32
#define RAD2    0.04f
#define NCH     64

// --------------------------------------------------------------------------
// Kernel 1: furthest point sampling. One 1024-thread block (32 waves) per
// batch. Points live in registers (8/thread); argmax via wave shuffle then
// a 32-entry LDS tree; new seed coords re-read as a uniform (scalar) load.
// --------------------------------------------------------------------------
__global__ __launch_bounds__(1024)
void fps_kernel(const float* __restrict__ xyz, float* __restrict__ child_xyz)
{
    __shared__ float red_v[32];
    __shared__ int   red_i[32];
    __shared__ int   s_last;
    __shared__ int   s_cidx[M_CHILD];

    const int b    = blockIdx.x;
    const int t    = threadIdx.x;
    const int lane = t & 31;
    const int wave = t >> 5;
    const float* base = xyz + (size_t)b * N_PTS * 3;

    float px[8], py[8], pz[8], md[8];
#pragma unroll
    for (int j = 0; j < 8; ++j) {
        int p = t + 1024 * j;
        px[j] = base[p * 3 + 0];
        py[j] = base[p * 3 + 1];
        pz[j] = base[p * 3 + 2];
        md[j] = 1e10f;
    }

    int last = 0;
    for (int i = 0; i < M_CHILD; ++i) {
        if (t == 0) s_cidx[i] = last;
        // uniform broadcast read of current seed point
        float lx = base[last * 3 + 0];
        float ly = base[last * 3 + 1];
        float lz = base[last * 3 + 2];
        float bv = -1.0f;
        int   bi = 0;
#pragma unroll
        for (int j = 0; j < 8; ++j) {
            float dx = px[j] - lx, dy = py[j] - ly, dz = pz[j] - lz;
            float d2 = dx * dx + dy * dy + dz * dz;
            md[j] = fminf(md[j], d2);
            if (md[j] > bv) { bv = md[j]; bi = t + 1024 * j; }
        }
#pragma unroll
        for (int off = 16; off > 0; off >>= 1) {
            float ov = __shfl_down(bv, off, 32);
            int   oi = __shfl_down(bi, off, 32);
            if (ov > bv || (ov == bv && oi < bi)) { bv = ov; bi = oi; }
        }
        if (lane == 0) { red_v[wave] = bv; red_i[wave] = bi; }
        __syncthreads();
        if (wave == 0) {
            float rv = red_v[lane];
            int   ri = red_i[lane];
#pragma unroll
            for (int off = 16; off > 0; off >>= 1) {
                float ov = __shfl_down(rv, off, 32);
                int   oi = __shfl_down(ri, off, 32);
                if (ov > rv || (ov == rv && oi < ri)) { rv = ov; ri = oi; }
            }
            if (lane == 0) s_last = ri;
        }
        __syncthreads();
        last = s_last;
    }
    __syncthreads();
    for (int i = t; i < M_CHILD; i += 1024) {
        int idx = s_cidx[i];
        child_xyz[((size_t)b * M_CHILD + i) * 3 + 0] = base[idx * 3 + 0];
        child_xyz[((size_t)b * M_CHILD + i) * 3 + 1] = base[idx * 3 + 1];
        child_xyz[((size_t)b * M_CHILD + i) * 3 + 2] = base[idx * 3 + 2];
    }
}

// --------------------------------------------------------------------------
// Kernel 2: ball query via V_WMMA_F32_16X16X4_F32.
//   D[i][j] = [px,py,pz,|p|^2] . [-2cx,-2cy,-2cz,1] = |p|^2 - 2 p.c
//   in-ball  <=>  D <= r^2 - |c|^2   (per-column threshold, one per lane)
// One wave owns 16 children (columns); loops point tiles of 16 rows in
// ascending index order -> first-K-within-radius semantics preserved.
// Tile loop unrolled x2: both ds_load_b64's issue before the wait, two
// back-to-back WMMAs per iteration; per-tile uniform guards keep the
// bit-extraction work proportional to actual hits.
// --------------------------------------------------------------------------
__global__ __launch_bounds__(256)
void ballquery_kernel(const float* __restrict__ xyz,
                      const float* __restrict__ child_xyz,
                      int* __restrict__ idx_out)
{
    __shared__ float4 spts[4096];   // 64 KB

    const int t    = threadIdx.x;
    const int lane = t & 31;
    const int wave = t >> 5;
    const int b    = blockIdx.x >> 5;           // 32 blocks per batch
    const int cblk = (blockIdx.x & 31) * 128;   // 128 children per block

    const float* pbase = xyz + (size_t)b * N_PTS * 3;

    const int jj    = lane & 15;
    const int child = cblk + wave * 16 + jj;
    const float cx = child_xyz[((size_t)b * M_CHILD + child) * 3 + 0];
    const float cy = child_xyz[((size_t)b * M_CHILD + child) * 3 + 1];
    const float cz = child_xyz[((size_t)b * M_CHILD + child) * 3 + 2];
    const float thr = RAD2 - (cx * cx + cy * cy + cz * cz);

    // B operand (4x16): VGPR0 = K0 / K2 per lane-half, VGPR1 = K1 / K3
    v2f bmat;
    bmat.x = (lane < 16) ? (-2.0f * cx) : (-2.0f * cz);
    bmat.y = (lane < 16) ? (-2.0f * cy) : 1.0f;

    int cnt = 0;
    int first = 0;
    const size_t rowbase = ((size_t)b * M_CHILD + child) * KNN;
    const float2* lds2 = (const float2*)spts;
    const int lslot = jj * 2 + (lane >> 4);     // per-lane A-operand slot

    for (int chunk = 0; chunk < 2; ++chunk) {
        __syncthreads();
        for (int q = 0; q < 16; ++q) {           // 4096 points per chunk
            int p  = t + q * 256;
            int gp = chunk * 4096 + p;
            float x = pbase[gp * 3 + 0];
            float y = pbase[gp * 3 + 1];
            float z = pbase[gp * 3 + 2];
            spts[p] = make_float4(x, y, z, x * x + y * y + z * z);
        }
        __syncthreads();

        for (int tile = 0; tile < 256; tile += 2) {
            const int tb0 = chunk * 4096 + tile * 16;
            // issue both LDS loads, then two WMMAs (shared B operand)
            float2 av0 = lds2[(size_t)tile * 32 + lslot];
            float2 av1 = lds2[(size_t)tile * 32 + 32 + lslot];
            v2f a0; a0.x = av0.x; a0.y = av0.y;
            v2f a1; a1.x = av1.x; a1.y = av1.y;
            v8f z8 = {0.f, 0.f, 0.f, 0.f, 0.f, 0.f, 0.f, 0.f};
            v8f acc0 = __builtin_amdgcn_wmma_f32_16x16x4_f32(
                false, a0, false, bmat, (short)0, z8, false, false);
            v8f acc1 = __builtin_amdgcn_wmma_f32_16x16x4_f32(
                false, a1, false, bmat, (short)0, z8, false, false);

            unsigned m0[8], m1[8];
#pragma unroll
            for (int r = 0; r < 8; ++r) m0[r] = (unsigned)__ballot(acc0[r] <= thr);
#pragma unroll
            for (int r = 0; r < 8; ++r) m1[r] = (unsigned)__ballot(acc1[r] <= thr);
            unsigned any0 = m0[0] | m0[1] | m0[2] | m0[3] | m0[4] | m0[5] | m0[6] | m0[7];
            unsigned any1 = m1[0] | m1[1] | m1[2] | m1[3] | m1[4] | m1[5] | m1[6] | m1[7];
            if ((any0 | any1) == 0) continue;    // uniform skip (common case)

            if (any0) {
                // per-lane 16-bit word: bit r -> point tb0+r (D VGPR r = rows r, r+8)
                unsigned wb = 0;
#pragma unroll
                for (int r = 0; r < 8; ++r) {
                    unsigned s = m0[r] >> jj;
                    wb |= (s & 1u) << r;
                    wb |= ((s >> 16) & 1u) << (r + 8);
                }
                if (lane >= 16) wb = 0;          // only lanes 0-15 own children
                while (__any(wb != 0)) {
                    if (wb) {
                        int bpos = __ffs(wb) - 1;
                        wb &= wb - 1u;
                        if (cnt < KNN) {
                            int pidx = tb0 + bpos;
                            if (cnt == 0) first = pidx;
                            idx_out[rowbase + cnt] = pidx;
                            ++cnt;
                        } else {
                            wb = 0;
                        }
                    }
                }
            }
            if (any1) {
                unsigned wb = 0;
#pragma unroll
                for (int r = 0; r < 8; ++r) {
                    unsigned s = m1[r] >> jj;
                    wb |= (s & 1u) << r;
                    wb |= ((s >> 16) & 1u) << (r + 8);
                }
                if (lane >= 16) wb = 0;
                while (__any(wb != 0)) {
                    if (wb) {
                        int bpos = __ffs(wb) - 1;
                        wb &= wb - 1u;
                        if (cnt < KNN) {
                            int pidx = tb0 + 16 + bpos;
                            if (cnt == 0) first = pidx;
                            idx_out[rowbase + cnt] = pidx;
                            ++cnt;
                        } else {
                            wb = 0;
                        }
                    }
                }
            }
        }
    }
    if (lane < 16) {                             // pad with first idx (0 if empty)
        for (int p = cnt; p < KNN; ++p) idx_out[rowbase + p] = first;
    }
}

// --------------------------------------------------------------------------
// Kernel 3: grouped feature max-pool. Block = 64 children x 64 channels;
// neighbor indices staged in LDS; gathers hit L2-resident feats rows.
// --------------------------------------------------------------------------
__global__ __launch_bounds__(256)
void groupmax_kernel(const float* __restrict__ feats,
                     const int* __restrict__ idx_in,
                     float* __restrict__ child_feats)
{
    __shared__ int sidx[64 * KNN];   // 8 KB
    const int t  = threadIdx.x;
    const int b  = blockIdx.x >> 6;          // 64 blocks per batch
    const int j0 = (blockIdx.x & 63) * 64;   // child tile base

    const size_t ibase = ((size_t)b * M_CHILD + j0) * KNN;
#pragma unroll
    for (int q = 0; q < 8; ++q)
        sidx[t + q * 256] = idx_in[ibase + t + q * 256];
    __syncthreads();

    const int c = t >> 2;
    const float* frow = feats + ((size_t)b * NCH + c) * N_PTS;
    __builtin_prefetch(frow, 0, 1);
    for (int q = 0; q < 16; ++q) {
        int jj = (t & 3) + (q << 2);
        const int* ip = &sidx[jj * KNN];
        float mx = -3.0e38f;
#pragma unroll
        for (int k = 0; k < KNN; ++k) mx = fmaxf(mx, frow[ip[k]]);
        child_feats[((size_t)b * NCH + c) * M_CHILD + (j0 + jj)] = mx;
    }
}

// --------------------------------------------------------------------------
extern "C" void kernel_launch(void* const* d_in, const int* in_sizes, int n_in,
                              void* d_out, int out_size, void* d_ws, size_t ws_size,
                              hipStream_t stream)
{
    (void)in_sizes; (void)n_in; (void)out_size; (void)ws_size;
    const float* xyz   = (const float*)d_in[0];   // (4, 8192, 3)
    const float* feats = (const float*)d_in[1];   // (4, 64, 8192)
    float* child_xyz   = (float*)d_out;                                   // (4,4096,3)
    float* child_feats = (float*)d_out + (size_t)BSZ * M_CHILD * 3;       // (4,64,4096)
    int*   idx_ws      = (int*)d_ws;              // 4*4096*32 ints = 2 MiB

    fps_kernel<<<BSZ, 1024, 0, stream>>>(xyz, child_xyz);
    ballquery_kernel<<<BSZ * (M_CHILD / 128), 256, 0, stream>>>(xyz, child_xyz, idx_ws);
    groupmax_kernel<<<BSZ * (M_CHILD / 64), 256, 0, stream>>>(feats, idx_ws, child_feats);
}